// ScalarGatedSelfAttention_18708877541816
// MI455X (gfx1250) — compile-verified
//
#include <hip/hip_runtime.h>

typedef __attribute__((ext_vector_type(2))) float v2f;
typedef __attribute__((ext_vector_type(8))) float v8f;

#define BB 16
#define HH 64
#define WW 64
#define CC 512
#define IC 64
#define ROWS (BB*HH*WW)   /* 65536 */
#define EPS 1e-3f

__device__ __forceinline__ v8f wmma_f32_k4(v2f a, v2f b, v8f c) {
  // D = A(16x4,f32) * B(4x16,f32) + C(16x16,f32)  -> v_wmma_f32_16x16x4_f32
  return __builtin_amdgcn_wmma_f32_16x16x4_f32(false, a, false, b, (short)0, c,
                                               false, false);
}

// butterfly add across the 16 lanes of each wave32 half via DS_SWIZZLE_B32
// group-of-32 xor mode: offset = (xor_mask<<10) | (or=0<<5) | (and_mask=0x1F)
template <int IMM>
__device__ __forceinline__ float swz_add(float v) {
  return v + __int_as_float(
      __builtin_amdgcn_ds_swizzle(__float_as_int(v), IMM));
}
__device__ __forceinline__ float hredux16(float v) {
  v = swz_add<(1 << 10) | 0x1F>(v);   // xor 1
  v = swz_add<(2 << 10) | 0x1F>(v);   // xor 2
  v = swz_add<(4 << 10) | 0x1F>(v);   // xor 4
  v = swz_add<(8 << 10) | 0x1F>(v);   // xor 8
  return v;
}

// ---------------------------------------------------------------------------
// Kernel 1: g = LN(x @ w_f), h = LN(x @ w_h)      (one wave per 16-row tile)
// ---------------------------------------------------------------------------
__global__ void __launch_bounds__(128)
k1_gemm_ln_gh(const float* __restrict__ x,
              const float* __restrict__ wf, const float* __restrict__ wh,
              const float* __restrict__ gf, const float* __restrict__ bf,
              const float* __restrict__ gh_, const float* __restrict__ bh,
              float* __restrict__ g, float* __restrict__ h)
{
  const int wave = threadIdx.x >> 5;
  const int lane = threadIdx.x & 31;
  const int m16  = lane & 15;
  const int half = lane >> 4;                  // 0: K 0/1, 1: K 2/3
  const int tile = blockIdx.x * 4 + wave;
  const int row0 = tile * 16;

  const float* xrow = x + (size_t)(row0 + m16) * CC + half * 2;

  v8f accg[4] = {};
  v8f acch[4] = {};

  for (int k = 0; k < CC; k += 4) {
    const float2 av = *(const float2*)(xrow + k);      // A frag: x[M][k+2h .. +1]
    v2f A; A.x = av.x; A.y = av.y;
    const int kr = k + half * 2;
#pragma unroll
    for (int nt = 0; nt < 4; ++nt) {
      const int col = nt * 16 + m16;                   // B frag: w[kr..kr+1][N]
      v2f Bf, Bh;
      Bf.x = wf[(size_t)kr * IC + col];
      Bf.y = wf[(size_t)(kr + 1) * IC + col];
      Bh.x = wh[(size_t)kr * IC + col];
      Bh.y = wh[(size_t)(kr + 1) * IC + col];
      accg[nt] = wmma_f32_k4(A, Bf, accg[nt]);
      acch[nt] = wmma_f32_k4(A, Bh, acch[nt]);
    }
  }

  // LayerNorm over IC=64: row M=j lives on lanes 0-15 (half 0) / M=j+8 on 16-31
#pragma unroll
  for (int j = 0; j < 8; ++j) {
    float s1g = 0.f, s2g = 0.f, s1h = 0.f, s2h = 0.f;
#pragma unroll
    for (int nt = 0; nt < 4; ++nt) {
      const float vg = accg[nt][j]; s1g += vg; s2g += vg * vg;
      const float vh = acch[nt][j]; s1h += vh; s2h += vh * vh;
    }
    s1g = hredux16(s1g); s2g = hredux16(s2g);
    s1h = hredux16(s1h); s2h = hredux16(s2h);
    const float mg = s1g * (1.0f / IC);
    const float rg = rsqrtf(s2g * (1.0f / IC) - mg * mg + EPS);
    const float mh = s1h * (1.0f / IC);
    const float rh = rsqrtf(s2h * (1.0f / IC) - mh * mh + EPS);
    const int row = row0 + j + half * 8;
#pragma unroll
    for (int nt = 0; nt < 4; ++nt) {
      const int n = nt * 16 + m16;
      g[(size_t)row * IC + n] = (accg[nt][j] - mg) * rg * gf[n]  + bf[n];
      h[(size_t)row * IC + n] = (acch[nt][j] - mh) * rh * gh_[n] + bh[n];
    }
  }
}

// ---------------------------------------------------------------------------
// Kernel 2: m[b,hh,w,:] = softmax_hh( g[b,w,hh,:] * g[b,hh,w,:] ) * h[b,hh,w,:]
// one block per (b, w); LDS-staged tiles
// ---------------------------------------------------------------------------
__global__ void __launch_bounds__(256)
k2_softmax(const float* __restrict__ g, const float* __restrict__ h,
           float* __restrict__ m)
{
  __shared__ float t1[HH * IC];   // g[b, w, hh, i]   (f, transposed gate)
  __shared__ float t2[HH * IC];   // g[b, hh, w, i]
  __shared__ float th[HH * IC];   // h[b, hh, w, i]
  __shared__ float redm[4 * IC];
  __shared__ float reds[4 * IC];

  const int b   = blockIdx.x >> 6;
  const int w   = blockIdx.x & 63;
  const int tid = threadIdx.x;

  const size_t base1 = (size_t)(b * HH + w) * WW * IC;   // contiguous 4096 floats
  for (int idx = tid; idx < HH * IC; idx += 256)
    t1[idx] = g[base1 + idx];

  const int i = tid & 63;
  const int c = tid >> 6;                                // 0..3 (16 hh each)
  for (int hh = c; hh < HH; hh += 4) {
    const size_t base = (size_t)((b * HH + hh) * WW + w) * IC;
    t2[hh * IC + i] = g[base + i];
    th[hh * IC + i] = h[base + i];
  }
  __syncthreads();

  float e[16];
  float lm = -3.402823466e38f;
#pragma unroll
  for (int jj = 0; jj < 16; ++jj) {
    const int hh = c * 16 + jj;
    const float s = t1[hh * IC + i] * t2[hh * IC + i];
    e[jj] = s;
    lm = fmaxf(lm, s);
  }
  redm[c * IC + i] = lm;
  __syncthreads();
  const float gm = fmaxf(fmaxf(redm[i], redm[IC + i]),
                         fmaxf(redm[2 * IC + i], redm[3 * IC + i]));
  float ls = 0.f;
#pragma unroll
  for (int jj = 0; jj < 16; ++jj) { e[jj] = __expf(e[jj] - gm); ls += e[jj]; }
  reds[c * IC + i] = ls;
  __syncthreads();
  const float inv = 1.0f / (reds[i] + reds[IC + i] + reds[2 * IC + i] + reds[3 * IC + i]);
#pragma unroll
  for (int jj = 0; jj < 16; ++jj) {
    const int hh = c * 16 + jj;
    const size_t base = (size_t)((b * HH + hh) * WW + w) * IC;
    m[base + i] = e[jj] * inv * th[hh * IC + i];
  }
}

// ---------------------------------------------------------------------------
// Kernel 3: out = x + LN(m @ w_fgh) * scale
// one block (4 waves) per 16-row tile; each wave owns 128 of 512 cols
// ---------------------------------------------------------------------------
__global__ void __launch_bounds__(128)
k3_out(const float* __restrict__ m, const float* __restrict__ wfgh,
       const float* __restrict__ gam, const float* __restrict__ bet,
       const float* __restrict__ scale, const float* __restrict__ x,
       float* __restrict__ out)
{
  __shared__ float lds_s[16 * 4];
  __shared__ float lds_q[16 * 4];
  __shared__ float lds_mean[16];
  __shared__ float lds_rs[16];

  const int wave = threadIdx.x >> 5;
  const int lane = threadIdx.x & 31;
  const int m16  = lane & 15;
  const int half = lane >> 4;
  const int row0 = blockIdx.x * 16;

  const float* mrow = m + (size_t)(row0 + m16) * IC + half * 2;

  v8f acc[8] = {};
  for (int k = 0; k < IC; k += 4) {
    const float2 av = *(const float2*)(mrow + k);
    v2f A; A.x = av.x; A.y = av.y;
    const int kr = k + half * 2;
#pragma unroll
    for (int nt = 0; nt < 8; ++nt) {
      const int col = wave * 128 + nt * 16 + m16;
      v2f Bv;
      Bv.x = wfgh[(size_t)kr * CC + col];
      Bv.y = wfgh[(size_t)(kr + 1) * CC + col];
      acc[nt] = wmma_f32_k4(A, Bv, acc[nt]);
    }
  }

  // per-row partial sums (this wave's 128 cols), then cross-wave via LDS
#pragma unroll
  for (int j = 0; j < 8; ++j) {
    float s1 = 0.f, s2 = 0.f;
#pragma unroll
    for (int nt = 0; nt < 8; ++nt) { const float v = acc[nt][j]; s1 += v; s2 += v * v; }
    s1 = hredux16(s1); s2 = hredux16(s2);
    if (m16 == 0) {
      const int r = j + half * 8;
      lds_s[r * 4 + wave] = s1;
      lds_q[r * 4 + wave] = s2;
    }
  }
  __syncthreads();
  if (threadIdx.x < 16) {
    const int r = threadIdx.x;
    const float s = lds_s[r * 4] + lds_s[r * 4 + 1] + lds_s[r * 4 + 2] + lds_s[r * 4 + 3];
    const float q = lds_q[r * 4] + lds_q[r * 4 + 1] + lds_q[r * 4 + 2] + lds_q[r * 4 + 3];
    const float mean = s * (1.0f / CC);
    const float var  = q * (1.0f / CC) - mean * mean;
    lds_mean[r] = mean;
    lds_rs[r]   = rsqrtf(var + EPS);
  }
  __syncthreads();

  const float sc = scale[0];
#pragma unroll
  for (int j = 0; j < 8; ++j) {
    const int r   = j + half * 8;
    const int row = row0 + r;
    const float mean = lds_mean[r];
    const float rs   = lds_rs[r];
#pragma unroll
    for (int nt = 0; nt < 8; ++nt) {
      const int n = wave * 128 + nt * 16 + m16;
      const float val = (acc[nt][j] - mean) * rs * gam[n] + bet[n];
      out[(size_t)row * CC + n] = x[(size_t)row * CC + n] + val * sc;
    }
  }
}

// ---------------------------------------------------------------------------
extern "C" void kernel_launch(void* const* d_in, const int* in_sizes, int n_in,
                              void* d_out, int out_size, void* d_ws, size_t ws_size,
                              hipStream_t stream) {
  const float* x     = (const float*)d_in[0];
  const float* wf    = (const float*)d_in[1];
  const float* wh    = (const float*)d_in[2];
  const float* wfgh  = (const float*)d_in[3];
  const float* gf    = (const float*)d_in[4];
  const float* bf    = (const float*)d_in[5];
  const float* gh    = (const float*)d_in[6];
  const float* bh    = (const float*)d_in[7];
  const float* gfgh  = (const float*)d_in[8];
  const float* bfgh  = (const float*)d_in[9];
  const float* scale = (const float*)d_in[10];
  float* out = (float*)d_out;

  float* gbuf = (float*)d_ws;                       // 16 MB
  float* hbuf = gbuf + (size_t)ROWS * IC;           // 16 MB
  float* mbuf = hbuf + (size_t)ROWS * IC;           // 16 MB

  k1_gemm_ln_gh<<<ROWS / 16 / 4, 128, 0, stream>>>(x, wf, wh, gf, bf, gh, bh,
                                                   gbuf, hbuf);
  k2_softmax<<<BB * WW, 256, 0, stream>>>(gbuf, hbuf, mbuf);
  k3_out<<<ROWS / 16, 128, 0, stream>>>(mbuf, wfgh, gfgh, bfgh, scale, x, out);
}